// ISDALoss_506806141617
// MI455X (gfx1250) — compile-verified
//
#include <hip/hip_runtime.h>
#include <math.h>

#define N_ROWS 4096
#define N_CLS  1000
#define N_CLSP 1024
#define N_FEAT 2048

typedef __attribute__((ext_vector_type(2))) float v2f;
typedef __attribute__((ext_vector_type(8))) float v8f;

// ---------------------------------------------------------------------------
// Kernel 1: zero the loss accumulator and build zero-padded W [1024 x 2048]
// ---------------------------------------------------------------------------
__global__ __launch_bounds__(256) void isda_prep_w(
    const float4* __restrict__ W4, float4* __restrict__ Wp4, float* __restrict__ loss0)
{
    int i = blockIdx.x * blockDim.x + threadIdx.x;
    if (i == 0) *loss0 = 0.0f;
    const int total4 = (N_CLSP * N_FEAT) / 4;   // 524288
    if (i >= total4) return;
    int c = i >> 9;                             // 512 float4 per row of 2048
    float4 v = make_float4(0.f, 0.f, 0.f, 0.f);
    if (c < N_CLS) v = W4[i];                   // same layout for c < 1000
    Wp4[i] = v;
}

// ---------------------------------------------------------------------------
// Kernel 2: t3[n] = sum_a W[y_n,a]^2 * cv[n,a]
// ---------------------------------------------------------------------------
__global__ __launch_bounds__(256) void isda_t3(
    const float* __restrict__ CV, const int* __restrict__ labels,
    const float* __restrict__ W, float* __restrict__ t3)
{
    const int n = blockIdx.x;
    const int lab = labels[n];
    const float* cvr = CV + (size_t)n * N_FEAT;
    const float* wr  = W  + (size_t)lab * N_FEAT;
    float s = 0.f;
    for (int a = threadIdx.x; a < N_FEAT; a += 256) {
        float w = wr[a];
        s += w * w * cvr[a];
    }
    __shared__ float red[256];
    red[threadIdx.x] = s;
    __syncthreads();
    for (int off = 128; off > 0; off >>= 1) {
        if (threadIdx.x < off) red[threadIdx.x] += red[threadIdx.x + off];
        __syncthreads();
    }
    if (threadIdx.x == 0) t3[n] = red[0];
}

// ---------------------------------------------------------------------------
// Kernel 3: fused triple-GEMM with V_WMMA_F32_16X16X4_F32.
// Each wave: 16 rows x 64 cols (4 c-tiles), accL = F*W, accD = u2*W^2 + u3*W
// with u2 = 0.5*ratio*cv, u3 = -ratio*cv*Wy (Wy gathered per-row on the fly).
// ---------------------------------------------------------------------------
__global__ __launch_bounds__(256) void isda_gemm(
    const float* __restrict__ F,  const float* __restrict__ CV,
    const int*   __restrict__ labels, const float* __restrict__ ratio_p,
    const float* __restrict__ W,  const float* __restrict__ Wpad,
    const float* __restrict__ bias, const float* __restrict__ t3,
    float* __restrict__ logits_out, float* __restrict__ aug_out)
{
    const float ratio = ratio_p[0];
    const float hr = 0.5f * ratio;
    const float nr = -ratio;

    const int lane = threadIdx.x & 31;
    const int wave = threadIdx.x >> 5;
    const int r16  = lane & 15;     // M (A-frag) / N (B,C,D frags)
    const int kg   = lane >> 4;     // k-pair group: cols 2*kg, 2*kg+1

    const int m0 = (blockIdx.y * 8 + wave) * 16;
    const int c0 = blockIdx.x * 64;

    const int arow = m0 + r16;
    const int lab  = labels[arow];

    const float* fp  = F    + (size_t)arow * N_FEAT + 2 * kg;
    const float* cvp = CV   + (size_t)arow * N_FEAT + 2 * kg;
    const float* wyp = W    + (size_t)lab  * N_FEAT + 2 * kg;
    const float* wp0 = Wpad + (size_t)(c0 + r16) * N_FEAT + 2 * kg;
    const float* wp1 = wp0 + 16 * N_FEAT;
    const float* wp2 = wp0 + 32 * N_FEAT;
    const float* wp3 = wp0 + 48 * N_FEAT;

    v8f accL[4];
    v8f accD[4];
#pragma unroll
    for (int t = 0; t < 4; ++t) { accL[t] = (v8f)0.f; accD[t] = (v8f)0.f; }

#pragma unroll 4
    for (int k = 0; k < N_FEAT; k += 4) {
        v2f fa  = *(const v2f*)(fp  + k);
        v2f ca  = *(const v2f*)(cvp + k);
        v2f wya = *(const v2f*)(wyp + k);
        v2f u2  = ca * hr;
        v2f u3  = ca * wya * nr;

        v2f b0 = *(const v2f*)(wp0 + k);
        v2f b1 = *(const v2f*)(wp1 + k);
        v2f b2 = *(const v2f*)(wp2 + k);
        v2f b3 = *(const v2f*)(wp3 + k);

        accL[0] = __builtin_amdgcn_wmma_f32_16x16x4_f32(false, fa, false, b0, (short)0, accL[0], false, false);
        accD[0] = __builtin_amdgcn_wmma_f32_16x16x4_f32(false, u2, false, b0 * b0, (short)0, accD[0], false, false);
        accD[0] = __builtin_amdgcn_wmma_f32_16x16x4_f32(false, u3, false, b0, (short)0, accD[0], false, false);

        accL[1] = __builtin_amdgcn_wmma_f32_16x16x4_f32(false, fa, false, b1, (short)0, accL[1], false, false);
        accD[1] = __builtin_amdgcn_wmma_f32_16x16x4_f32(false, u2, false, b1 * b1, (short)0, accD[1], false, false);
        accD[1] = __builtin_amdgcn_wmma_f32_16x16x4_f32(false, u3, false, b1, (short)0, accD[1], false, false);

        accL[2] = __builtin_amdgcn_wmma_f32_16x16x4_f32(false, fa, false, b2, (short)0, accL[2], false, false);
        accD[2] = __builtin_amdgcn_wmma_f32_16x16x4_f32(false, u2, false, b2 * b2, (short)0, accD[2], false, false);
        accD[2] = __builtin_amdgcn_wmma_f32_16x16x4_f32(false, u3, false, b2, (short)0, accD[2], false, false);

        accL[3] = __builtin_amdgcn_wmma_f32_16x16x4_f32(false, fa, false, b3, (short)0, accL[3], false, false);
        accD[3] = __builtin_amdgcn_wmma_f32_16x16x4_f32(false, u2, false, b3 * b3, (short)0, accD[3], false, false);
        accD[3] = __builtin_amdgcn_wmma_f32_16x16x4_f32(false, u3, false, b3, (short)0, accD[3], false, false);
    }

    // Epilogue. D layout: VGPR r -> row m0 + r + 8*kg, lane%16 -> column.
    float t3v[8];
#pragma unroll
    for (int r = 0; r < 8; ++r) t3v[r] = hr * t3[m0 + r + 8 * kg];

#pragma unroll
    for (int t = 0; t < 4; ++t) {
        const int cc = c0 + 16 * t + r16;
        const bool valid = (cc < N_CLS);
        const float bs = bias[valid ? cc : 0];
#pragma unroll
        for (int r = 0; r < 8; ++r) {
            const int row = m0 + r + 8 * kg;
            const float lg = accL[t][r] + bs;
            const float ag = lg + accD[t][r] + t3v[r];
            if (valid) {
                logits_out[(size_t)row * N_CLS + cc] = lg;
                aug_out[(size_t)row * N_CLSP + cc]   = ag;
            }
        }
    }
}

// ---------------------------------------------------------------------------
// Kernel 4: per-row log-sum-exp over aug logits, accumulate mean NLL.
// ---------------------------------------------------------------------------
__global__ __launch_bounds__(256) void isda_loss(
    const float* __restrict__ aug, const int* __restrict__ labels,
    float* __restrict__ loss_out)
{
    const int n = blockIdx.x;
    const float* row = aug + (size_t)n * N_CLSP;
    const int tid = threadIdx.x;

    __shared__ float red[256];

    float mx = -INFINITY;
    for (int c = tid; c < N_CLS; c += 256) mx = fmaxf(mx, row[c]);
    red[tid] = mx;
    __syncthreads();
    for (int off = 128; off > 0; off >>= 1) {
        if (tid < off) red[tid] = fmaxf(red[tid], red[tid + off]);
        __syncthreads();
    }
    mx = red[0];
    __syncthreads();

    float s = 0.f;
    for (int c = tid; c < N_CLS; c += 256) s += expf(row[c] - mx);
    red[tid] = s;
    __syncthreads();
    for (int off = 128; off > 0; off >>= 1) {
        if (tid < off) red[tid] += red[tid + off];
        __syncthreads();
    }

    if (tid == 0) {
        const float lse = mx + logf(red[0]);
        const float nll = lse - row[labels[n]];
        atomicAdd(loss_out, nll * (1.0f / (float)N_ROWS));
    }
}

// ---------------------------------------------------------------------------
extern "C" void kernel_launch(void* const* d_in, const int* in_sizes, int n_in,
                              void* d_out, int out_size, void* d_ws, size_t ws_size,
                              hipStream_t stream)
{
    const float* F      = (const float*)d_in[0];
    const int*   labels = (const int*)  d_in[1];
    const float* CV     = (const float*)d_in[2];
    const float* ratio  = (const float*)d_in[3];
    const float* W      = (const float*)d_in[4];
    const float* bias   = (const float*)d_in[5];

    float* out = (float*)d_out;            // out[0] = loss, out+1 = logits [N, 1000]

    char* ws = (char*)d_ws;
    float* t3   = (float*)ws;                                          // 16 KB
    float* aug  = (float*)(ws + 16384);                                // 16 MB
    float* Wpad = (float*)(ws + 16384 + (size_t)N_ROWS * N_CLSP * 4);  // 8 MB

    // 1) pad W, zero loss accumulator
    {
        const int total4 = (N_CLSP * N_FEAT) / 4;
        isda_prep_w<<<(total4 + 255) / 256, 256, 0, stream>>>(
            (const float4*)W, (float4*)Wpad, out);
    }
    // 2) t3 row reduction
    isda_t3<<<N_ROWS, 256, 0, stream>>>(CV, labels, W, t3);
    // 3) fused WMMA GEMM
    {
        dim3 grid(N_CLSP / 64, N_ROWS / (16 * 8));   // (16, 32), 8 waves/block
        isda_gemm<<<grid, 256, 0, stream>>>(F, CV, labels, ratio, W, Wpad,
                                            bias, t3, out + 1, aug);
    }
    // 4) softmax CE loss
    isda_loss<<<N_ROWS, 256, 0, stream>>>(aug, labels, out);
}